// Encoder_59365037966048
// MI455X (gfx1250) — compile-verified
//
#include <hip/hip_runtime.h>

typedef float v2f __attribute__((ext_vector_type(2)));
typedef float v8f __attribute__((ext_vector_type(8)));

#define KCHUNK 32   // K elements staged to LDS per block iteration
#define MT 4        // 16x16 M-tiles per wave (64 rows per wave)
#define MAXK 4096   // max Cin*KH*KW in this network (256*4*4)

#if defined(__has_builtin)
#if __has_builtin(__builtin_amdgcn_tensor_load_to_lds) && \
    __has_builtin(__builtin_amdgcn_s_wait_tensorcnt)
#define USE_TDM 1
#endif
#endif

#ifdef USE_TDM
typedef unsigned int v4u __attribute__((ext_vector_type(4)));
typedef int v4i __attribute__((ext_vector_type(4)));
typedef int v8i __attribute__((ext_vector_type(8)));

// ---------------------------------------------------------------------------
// Tensor Data Mover: DMA a 2D tile (32 contiguous elems x 32 rows) into LDS.
// D# per CDNA5 ISA ch.8: data_size=4B; pad 1 dword every 32 dwords => LDS row
// stride 33 (bank-conflict free); tensor_dim clamps with hardware zero-fill.
// ---------------------------------------------------------------------------
__device__ __forceinline__ void tdm_load_tile(const float* gsrc,
                                              unsigned lds_byte, int dim0,
                                              int dim1, long row_stride) {
  unsigned long long ga = (unsigned long long)gsrc;
  int d0 = dim0 < 0 ? 0 : dim0;
  int d1 = dim1 < 0 ? 0 : dim1;
  v4u g0;
  g0[0] = 1u;                                   // count=1 (valid user D#)
  g0[1] = lds_byte;                             // lds_addr
  g0[2] = (unsigned)(ga & 0xffffffffu);         // global_addr[31:0]
  g0[3] = (unsigned)((ga >> 32) & 0x01ffffffu)  // global_addr[56:32]
          | 0x80000000u;                        // type=2 ("image")
  v8i g1;
  g1[0] = (2 << 16)      // data_size = 4 bytes
          | (1 << 20)    // pad_enable
          | (4 << 22);   // pad_interval: 32 dwords
  g1[1] = (d0 & 0xffff) << 16;                        // tensor_dim0[15:0]
  g1[2] = (int)(((unsigned)d0 >> 16) | ((unsigned)(d1 & 0xffff) << 16));
  g1[3] = (int)(((unsigned)d1 >> 16) | ((unsigned)KCHUNK << 16));  // tile_dim0
  g1[4] = 32;                                          // tile_dim1 (tile_dim2=0)
  g1[5] = (int)(row_stride & 0xffffffff);              // dim0_stride[31:0]
  g1[6] = (int)((row_stride >> 32) & 0xffff);          // dim0_stride[47:32]
  g1[7] = 0;
  v4i zz = {0, 0, 0, 0};
#if __clang_major__ >= 23
  v8i z8 = {0, 0, 0, 0, 0, 0, 0, 0};
  __builtin_amdgcn_tensor_load_to_lds(g0, g1, zz, zz, z8, 0);
#else
  __builtin_amdgcn_tensor_load_to_lds(g0, g1, zz, zz, 0);
#endif
}
#endif  // USE_TDM

// ---------------------------------------------------------------------------
// WMMA core: D = A(16x4 f32) * B(4x16 f32) + C(16x16 f32), wave32.
// ---------------------------------------------------------------------------
__device__ __forceinline__ v8f wmma4(v2f a, v2f b, v8f c) {
  return __builtin_amdgcn_wmma_f32_16x16x4_f32(false, a, false, b, (short)0, c,
                                               false, false);
}

// ---------------------------------------------------------------------------
// GroupNorm statistics: one block per (batch, group); deterministic tree reduce
// ---------------------------------------------------------------------------
__global__ void __launch_bounds__(256) gn_stats_kernel(
    const float* __restrict__ x, float* __restrict__ stats, int C, int HW) {
  int bg = blockIdx.x;
  int b = bg >> 5, g = bg & 31;
  int cpg = C / 32;
  long base = ((long)b * C + (long)g * cpg) * HW;
  long n = (long)cpg * HW;
  float s = 0.f, s2 = 0.f;
  for (long i = threadIdx.x; i < n; i += 256) {
    float v = x[base + i];
    s += v; s2 += v * v;
  }
  __shared__ float sh[256], sh2[256];
  sh[threadIdx.x] = s; sh2[threadIdx.x] = s2;
  __syncthreads();
  for (int o = 128; o > 0; o >>= 1) {
    if (threadIdx.x < o) {
      sh[threadIdx.x] += sh[threadIdx.x + o];
      sh2[threadIdx.x] += sh2[threadIdx.x + o];
    }
    __syncthreads();
  }
  if (threadIdx.x == 0) {
    float mean = sh[0] / (float)n;
    float var = sh2[0] / (float)n - mean * mean;
    stats[bg * 2] = mean;
    stats[bg * 2 + 1] = rsqrtf(var + 1e-5f);
  }
}

__global__ void __launch_bounds__(256) gn_apply_kernel(
    const float* __restrict__ x, const float* __restrict__ stats,
    const float* __restrict__ sc, const float* __restrict__ bi,
    float* __restrict__ y, int C, int HW, long total, int do_silu) {
  long i = (long)blockIdx.x * blockDim.x + threadIdx.x;
  long stride = (long)gridDim.x * blockDim.x;
  int cpg = C / 32;
  for (; i < total; i += stride) {
    long c_ = (i / HW) % C;
    long b_ = i / ((long)C * HW);
    int bg = (int)(b_ * 32 + c_ / cpg);
    float v = (x[i] - stats[bg * 2]) * stats[bg * 2 + 1] * sc[c_] + bi[c_];
    if (do_silu) v = v * (1.f / (1.f + __expf(-v)));
    y[i] = v;
  }
}

// ---------------------------------------------------------------------------
// Implicit-GEMM convolution: M = Hout*Wout, N = Cout, K = Cin*KH*KW.
// Weight chunk staged to LDS by the Tensor Data Mover (NK layout, stride 33);
// im2col offset tables in LDS keep the K loop division-free.
// grid = (ceil(M/256), ceil(Cout/32), B)
// ---------------------------------------------------------------------------
__global__ void __launch_bounds__(256) conv_wmma_kernel(
    const float* __restrict__ X, const float* __restrict__ W,
    const float* __restrict__ bias, const float* __restrict__ resid,
    float* __restrict__ Y, int Cin, int Hin, int Win, int Cout, int Hout,
    int Wout, int KH, int KW, int stride, int pad) {
  __shared__ float Bsh[KCHUNK * 33];
  __shared__ int Koff[MAXK];
  __shared__ int Kpos[MAXK];

  const int Mout = Hout * Wout;
  const int Ktot = Cin * KH * KW;
  const int KHW = KH * KW;
  const int wv = threadIdx.x >> 5;
  const int lane = threadIdx.x & 31;
  const int half = lane >> 4;
  const int l16 = lane & 15;
  const int b = blockIdx.z;
  const int nblk = blockIdx.y * 32;
  const int ncol = (wv >> 2) * 16 + l16;  // 0..31 within block N-span
  const int co = nblk + ncol;
  const bool cok = co < Cout;
  const int mbase = (blockIdx.x * 4 + (wv & 3)) * (MT * 16);

  // im2col tables (divisions done once per k, not per iteration)
  for (int k = threadIdx.x; k < Ktot; k += 256) {
    int ci = k / KHW;
    int r2 = k - ci * KHW;
    int kh = r2 / KW;
    int kw = r2 - kh * KW;
    Koff[k] = (ci * Hin + kh) * Win + kw;
    Kpos[k] = (kh << 16) | kw;
  }

  // per-lane, per-tile output pixel decode
  int ihb[MT], iwb[MT];
  bool mok[MT];
  long arow[MT];
  const long xbase = (long)b * Cin * Hin * Win;
#pragma unroll
  for (int t = 0; t < MT; ++t) {
    int am = mbase + t * 16 + l16;
    mok[t] = am < Mout;
    int aoh = am / Wout, aow = am - aoh * Wout;
    ihb[t] = aoh * stride - pad;
    iwb[t] = aow * stride - pad;
    arow[t] = xbase + (long)ihb[t] * Win + iwb[t];
  }

#ifdef USE_TDM
  const unsigned ldsB = (unsigned)(unsigned long long)(void*)Bsh;
#else
  const int sn = threadIdx.x >> 3;
  const int skq = (threadIdx.x & 7) * 4;
  const int sco = nblk + sn;
  const bool scok = sco < Cout;
  const long wrow = (long)sco * Ktot;
#endif

  v8f z = {0.f, 0.f, 0.f, 0.f, 0.f, 0.f, 0.f, 0.f};
  v8f acc[MT];
#pragma unroll
  for (int t = 0; t < MT; ++t) acc[t] = z;

  for (int kk0 = 0; kk0 < Ktot; kk0 += KCHUNK) {
    __syncthreads();  // tile consumed (and tables ready on first pass)
#ifdef USE_TDM
    if (wv == 0) {
      // NK tile: dim0 = k (contiguous in OIHW weights), dim1 = cout rows
      tdm_load_tile(W + (long)nblk * Ktot + kk0, ldsB, Ktot - kk0,
                    Cout - nblk, Ktot);
      __builtin_amdgcn_s_wait_tensorcnt(0);
    } else if (wv == 1 && kk0 + KCHUNK < Ktot && nblk + lane < Cout) {
      __builtin_prefetch(&W[(long)(nblk + lane) * Ktot + kk0 + KCHUNK], 0, 1);
    }
#else
#pragma unroll
    for (int j = 0; j < 4; ++j) {
      int k = kk0 + skq + j;
      float v = (scok && k < Ktot) ? W[wrow + k] : 0.f;
      Bsh[sn * 33 + skq + j] = v;  // NK layout, same as TDM path
    }
    if (scok && kk0 + KCHUNK + skq < Ktot)
      __builtin_prefetch(&W[wrow + kk0 + KCHUNK + skq], 0, 1);
#endif
    __syncthreads();

    const int klim = (Ktot - kk0 < KCHUNK) ? (Ktot - kk0) : KCHUNK;
    for (int k4 = 0; k4 < klim; k4 += 4) {
      const int ka = k4 + half * 2;
      v2f bf;
      bf.x = Bsh[ncol * 33 + ka];      // NK layout
      bf.y = Bsh[ncol * 33 + ka + 1];
#pragma unroll
      for (int t = 0; t < MT; ++t) {
        v2f af;
#pragma unroll
        for (int u = 0; u < 2; ++u) {
          int kg = kk0 + ka + u;
          float av = 0.f;
          if (kg < Ktot && mok[t]) {
            int pos = Kpos[kg];
            int ih = ihb[t] + (pos >> 16);
            int iw = iwb[t] + (pos & 0xffff);
            if ((unsigned)ih < (unsigned)Hin && (unsigned)iw < (unsigned)Win)
              av = X[arow[t] + Koff[kg]];
          }
          if (u == 0) af.x = av; else af.y = av;
        }
        acc[t] = wmma4(af, bf, acc[t]);
      }
    }
  }

  float bb = cok ? bias[co] : 0.f;
  const long ybase = (long)b * Cout * Mout;
#pragma unroll
  for (int t = 0; t < MT; ++t) {
#pragma unroll
    for (int r = 0; r < 8; ++r) {
      int mi = mbase + t * 16 + r + 8 * half;
      if (mi < Mout && cok) {
        long oi = ybase + (long)co * Mout + mi;
        float v = acc[t][r] + bb;
        if (resid) v += resid[oi];
        Y[oi] = v;
      }
    }
  }
}

// ---------------------------------------------------------------------------
// Generic strided batched GEMM: C = alpha*A*B + bias + resid.
// B chunk staged by TDM: NK tile when sbk==1, KN tile when sbn==1 (every GEMM
// in this net has one unit stride). grid = (ceil(M/256), ceil(N/32), z)
// ---------------------------------------------------------------------------
__global__ void __launch_bounds__(256) gemm_wmma_kernel(
    const float* __restrict__ A, const float* __restrict__ Bm,
    const float* __restrict__ bias, const float* __restrict__ R,
    float* __restrict__ C, int M, int N, int K, long sam, long sak, long sbk,
    long sbn, long scm, long scn, int zInner, long aBI, long aBO, long bBI,
    long bBO, long cBI, long cBO, long rBI, long rBO, float alpha) {
  __shared__ float Bsh[KCHUNK * 33];
  const int wv = threadIdx.x >> 5;
  const int lane = threadIdx.x & 31;
  const int half = lane >> 4;
  const int l16 = lane & 15;
  const int nblk = blockIdx.y * 32;
  const int ncol = (wv >> 2) * 16 + l16;
  const int bn = nblk + ncol;
  const bool bnok = bn < N;
  const int mbase = (blockIdx.x * 4 + (wv & 3)) * (MT * 16);

  const int bz = blockIdx.z;
  const int zi = bz % zInner, zo = bz / zInner;
  const float* Ab = A + zi * aBI + zo * aBO;
  const float* Bb = Bm + zi * bBI + zo * bBO;
  float* Cb = C + zi * cBI + zo * cBO;
  const float* Rb = R ? (R + zi * rBI + zo * rBO) : (const float*)0;

  int am[MT];
  bool mok[MT];
#pragma unroll
  for (int t = 0; t < MT; ++t) {
    am[t] = mbase + t * 16 + l16;
    mok[t] = am[t] < M;
  }

#ifdef USE_TDM
  const bool layNK = (sbk == 1);
  const unsigned ldsB = (unsigned)(unsigned long long)(void*)Bsh;
#else
  const bool layNK = false;
  const int sn = threadIdx.x >> 3;
  const int skq = (threadIdx.x & 7) * 4;
  const bool snok = (nblk + sn) < N;
  const long sbase = (long)(nblk + sn) * sbn;
#endif

  v8f z = {0.f, 0.f, 0.f, 0.f, 0.f, 0.f, 0.f, 0.f};
  v8f acc[MT];
#pragma unroll
  for (int t = 0; t < MT; ++t) acc[t] = z;

  for (int kk0 = 0; kk0 < K; kk0 += KCHUNK) {
    __syncthreads();
#ifdef USE_TDM
    if (wv == 0) {
      const float* ts = layNK ? (Bb + kk0 + (long)nblk * sbn)
                              : (Bb + nblk + (long)kk0 * sbk);
      int d0 = layNK ? (K - kk0) : (N - nblk);
      int d1 = layNK ? (N - nblk) : (K - kk0);
      tdm_load_tile(ts, ldsB, d0, d1, layNK ? sbn : sbk);
      __builtin_amdgcn_s_wait_tensorcnt(0);
    } else if (wv == 1 && kk0 + KCHUNK < K) {
      if (layNK) {
        if (nblk + lane < N)
          __builtin_prefetch(&Bb[kk0 + KCHUNK + (long)(nblk + lane) * sbn], 0, 1);
      } else {
        if (kk0 + KCHUNK + lane < K)
          __builtin_prefetch(&Bb[nblk + (long)(kk0 + KCHUNK + lane) * sbk], 0, 1);
      }
    }
#else
#pragma unroll
    for (int j = 0; j < 4; ++j) {
      int k = kk0 + skq + j;
      float v = (snok && k < K) ? Bb[(long)k * sbk + sbase] : 0.f;
      Bsh[(skq + j) * 33 + sn] = v;  // KN layout
    }
    if (snok && kk0 + KCHUNK + skq < K)
      __builtin_prefetch(&Bb[(long)(kk0 + KCHUNK + skq) * sbk + sbase], 0, 1);
#endif
    __syncthreads();

    const int klim = (K - kk0 < KCHUNK) ? (K - kk0) : KCHUNK;
    for (int k4 = 0; k4 < klim; k4 += 4) {
      const int ka = k4 + half * 2;
      v2f bf;
      bf.x = layNK ? Bsh[ncol * 33 + ka] : Bsh[ka * 33 + ncol];
      bf.y = layNK ? Bsh[ncol * 33 + ka + 1] : Bsh[(ka + 1) * 33 + ncol];
#pragma unroll
      for (int t = 0; t < MT; ++t) {
        v2f af;
#pragma unroll
        for (int u = 0; u < 2; ++u) {
          int kg = kk0 + ka + u;
          float av = (mok[t] && kg < K)
                         ? Ab[(long)am[t] * sam + (long)kg * sak]
                         : 0.f;
          if (u == 0) af.x = av; else af.y = av;
        }
        acc[t] = wmma4(af, bf, acc[t]);
      }
    }
  }

  float bb = (bias && bnok) ? bias[bn] : 0.f;
#pragma unroll
  for (int t = 0; t < MT; ++t) {
#pragma unroll
    for (int r = 0; r < 8; ++r) {
      int mi = mbase + t * 16 + r + 8 * half;
      if (mi < M && bnok) {
        long oi = (long)mi * scm + (long)bn * scn;
        float v = acc[t][r] * alpha + bb;
        if (Rb) v += Rb[oi];
        Cb[oi] = v;
      }
    }
  }
}

// ---------------------------------------------------------------------------
// Row softmax (in place), one block per row
// ---------------------------------------------------------------------------
__global__ void __launch_bounds__(256) softmax_rows_kernel(float* __restrict__ S,
                                                           int N) {
  long row = blockIdx.x;
  float* p = S + row * (long)N;
  __shared__ float sh[256];
  float mx = -3.4e38f;
  for (int i = threadIdx.x; i < N; i += 256) mx = fmaxf(mx, p[i]);
  sh[threadIdx.x] = mx; __syncthreads();
  for (int o = 128; o > 0; o >>= 1) {
    if (threadIdx.x < o) sh[threadIdx.x] = fmaxf(sh[threadIdx.x], sh[threadIdx.x + o]);
    __syncthreads();
  }
  mx = sh[0]; __syncthreads();
  float sum = 0.f;
  for (int i = threadIdx.x; i < N; i += 256) {
    float e = __expf(p[i] - mx);
    p[i] = e; sum += e;
  }
  sh[threadIdx.x] = sum; __syncthreads();
  for (int o = 128; o > 0; o >>= 1) {
    if (threadIdx.x < o) sh[threadIdx.x] += sh[threadIdx.x + o];
    __syncthreads();
  }
  float inv = 1.f / sh[0];
  for (int i = threadIdx.x; i < N; i += 256) p[i] *= inv;
}

// ---------------------------------------------------------------------------
// Vector quantize: one block per point (B*HW=8192), scan 8192 codes (C=3)
// ---------------------------------------------------------------------------
__global__ void __launch_bounds__(256) vq_assign_kernel(
    const float* __restrict__ z, const float* __restrict__ cb,
    float* __restrict__ qout, float* __restrict__ sqerr, int Ncodes) {
  int p = blockIdx.x;
  int b = p >> 10, n = p & 1023;
  long zb = (long)b * 3072 + n;
  float x0 = z[zb], x1 = z[zb + 1024], x2 = z[zb + 2048];
  float best = 3.4e38f; int bi = 0;
  for (int c = threadIdx.x; c < Ncodes; c += 256) {
    float d0 = cb[c * 3] - x0, d1 = cb[c * 3 + 1] - x1, d2 = cb[c * 3 + 2] - x2;
    float d = d0 * d0 + d1 * d1 + d2 * d2;
    if (d < best) { best = d; bi = c; }
  }
  __shared__ float shd[256];
  __shared__ int shi[256];
  shd[threadIdx.x] = best; shi[threadIdx.x] = bi;
  __syncthreads();
  for (int o = 128; o > 0; o >>= 1) {
    if (threadIdx.x < o) {
      float od = shd[threadIdx.x + o]; int oi = shi[threadIdx.x + o];
      if (od < shd[threadIdx.x] || (od == shd[threadIdx.x] && oi < shi[threadIdx.x])) {
        shd[threadIdx.x] = od; shi[threadIdx.x] = oi;
      }
    }
    __syncthreads();
  }
  if (threadIdx.x == 0) {
    int c = shi[0];
    float q0 = cb[c * 3], q1 = cb[c * 3 + 1], q2 = cb[c * 3 + 2];
    qout[zb] = q0; qout[zb + 1024] = q1; qout[zb + 2048] = q2;
    float e0 = q0 - x0, e1 = q1 - x1, e2 = q2 - x2;
    sqerr[p] = e0 * e0 + e1 * e1 + e2 * e2;
  }
}

__global__ void __launch_bounds__(256) vq_loss_kernel(
    const float* __restrict__ sqerr, float* __restrict__ out2, int n,
    float invTotal) {
  __shared__ float sh[256];
  float s = 0.f;
  for (int i = threadIdx.x; i < n; i += 256) s += sqerr[i];
  sh[threadIdx.x] = s; __syncthreads();
  for (int o = 128; o > 0; o >>= 1) {
    if (threadIdx.x < o) sh[threadIdx.x] += sh[threadIdx.x + o];
    __syncthreads();
  }
  if (threadIdx.x == 0) { float L = sh[0] * invTotal; out2[0] = L; out2[1] = L; }
}

// ===========================================================================
// Host side
// ===========================================================================
struct RnP {
  const float *gn1_s, *gn1_b, *c1_w, *c1_b, *gn2_s, *gn2_b, *c2_w, *c2_b,
      *rc_w, *rc_b, *ds_w, *ds_b;
};
struct Params {
  const float *x, *conv_in_w, *conv_in_b;
  RnP down[3];
  RnP rn1, rn2;
  const float *agn_s, *agn_b, *qkv_w, *qkv_b, *out_w, *out_b;
  const float *no_s, *no_b, *conv_out_w, *conv_out_b, *pq_w, *pq_b, *codebook;
};

static void fill_rn(RnP& r, const float* const* F, int& i, bool sorted, bool ds) {
  if (!sorted) {
    r.gn1_s = F[i++]; r.gn1_b = F[i++]; r.c1_w = F[i++]; r.c1_b = F[i++];
    r.gn2_s = F[i++]; r.gn2_b = F[i++]; r.c2_w = F[i++]; r.c2_b = F[i++];
    r.rc_w = F[i++]; r.rc_b = F[i++];
    if (ds) { r.ds_w = F[i++]; r.ds_b = F[i++]; }
  } else {
    r.c1_b = F[i++]; r.c1_w = F[i++]; r.c2_b = F[i++]; r.c2_w = F[i++];
    if (ds) { r.ds_b = F[i++]; r.ds_w = F[i++]; }
    r.gn1_b = F[i++]; r.gn1_s = F[i++]; r.gn2_b = F[i++]; r.gn2_s = F[i++];
    r.rc_b = F[i++]; r.rc_w = F[i++];
  }
}

static Params map_params(const float* const* F, const int* sz) {
  Params P{};
  bool insertion = (sz[0] == 8 * 3 * 256 * 256);  // x first => insertion order
  int i = 0;
  if (insertion) {
    P.x = F[i++]; P.conv_in_w = F[i++]; P.conv_in_b = F[i++];
    for (int d = 0; d < 3; ++d) fill_rn(P.down[d], F, i, false, true);
    fill_rn(P.rn1, F, i, false, false);
    fill_rn(P.rn2, F, i, false, false);
    P.agn_s = F[i++]; P.agn_b = F[i++]; P.qkv_w = F[i++]; P.qkv_b = F[i++];
    P.out_w = F[i++]; P.out_b = F[i++];
    P.no_s = F[i++]; P.no_b = F[i++];
    P.conv_out_w = F[i++]; P.conv_out_b = F[i++];
    P.pq_w = F[i++]; P.pq_b = F[i++];
    P.codebook = F[i++];
  } else {  // JAX pytree sorted-key order: params(sorted) then x
    P.codebook = F[i++];
    P.conv_in_b = F[i++]; P.conv_in_w = F[i++];
    P.conv_out_b = F[i++]; P.conv_out_w = F[i++];
    for (int d = 0; d < 3; ++d) fill_rn(P.down[d], F, i, true, true);
    P.agn_b = F[i++]; P.agn_s = F[i++]; P.out_b = F[i++]; P.out_w = F[i++];
    P.qkv_b = F[i++]; P.qkv_w = F[i++];
    fill_rn(P.rn1, F, i, true, false);
    fill_rn(P.rn2, F, i, true, false);
    P.no_b = F[i++]; P.no_s = F[i++];
    P.pq_b = F[i++]; P.pq_w = F[i++];
    P.x = F[i++];
  }
  return P;
}

static void run_gn(const float* x, float* y, const float* s, const float* b,
                   int B, int C, int HW, int silu, float* stats,
                   hipStream_t st) {
  gn_stats_kernel<<<B * 32, 256, 0, st>>>(x, stats, C, HW);
  long total = (long)B * C * HW;
  long want = (total + 255) / 256;
  int blocks = (int)(want > 8192 ? 8192 : want);
  gn_apply_kernel<<<blocks, 256, 0, st>>>(x, stats, s, b, y, C, HW, total, silu);
}

static void run_conv(const float* X, const float* W, const float* bias,
                     const float* resid, float* Y, int B, int Cin, int Hin,
                     int Win, int Cout, int KH, int KW, int stride, int pad,
                     hipStream_t st) {
  int Hout = (Hin + 2 * pad - KH) / stride + 1;
  int Wout = (Win + 2 * pad - KW) / stride + 1;
  int Mout = Hout * Wout;
  dim3 g((Mout + 255) / 256, (Cout + 31) / 32, B);
  conv_wmma_kernel<<<g, 256, 0, st>>>(X, W, bias, resid, Y, Cin, Hin, Win,
                                      Cout, Hout, Wout, KH, KW, stride, pad);
}

static void run_gemm(const float* A, const float* Bm, const float* bias,
                     const float* R, float* C, int M, int N, int K, long sam,
                     long sak, long sbk, long sbn, long scm, long scn,
                     int zInner, int zOuter, long aBI, long aBO, long bBI,
                     long bBO, long cBI, long cBO, long rBI, long rBO,
                     float alpha, hipStream_t st) {
  dim3 g((M + 255) / 256, (N + 31) / 32, zInner * zOuter);
  gemm_wmma_kernel<<<g, 256, 0, st>>>(A, Bm, bias, R, C, M, N, K, sam, sak,
                                      sbk, sbn, scm, scn, zInner, aBI, aBO,
                                      bBI, bBO, cBI, cBO, rBI, rBO, alpha);
}

// in-place resnet block: result lands back in io
static void resnet(const RnP& p, float* io, float* t1, float* t2, int B,
                   int Cin, int Cout, int H, int W, float* stats,
                   hipStream_t st) {
  int HW = H * W;
  run_gn(io, t1, p.gn1_s, p.gn1_b, B, Cin, HW, 1, stats, st);
  run_conv(t1, p.c1_w, p.c1_b, nullptr, t2, B, Cin, H, W, Cout, 3, 3, 1, 1, st);
  run_gn(t2, t1, p.gn2_s, p.gn2_b, B, Cout, HW, 1, stats, st);
  // 1x1 shortcut conv: rc(io) -> t2 (h1 dead)
  run_gemm(io, p.rc_w, p.rc_b, nullptr, t2, HW, Cout, Cin, 1, HW, 1, Cin, 1,
           HW, B, 1, (long)Cin * HW, 0, 0, 0, (long)Cout * HW, 0, 0, 0, 1.f,
           st);
  // out = conv(t1) + rc(t2) -> io
  run_conv(t1, p.c2_w, p.c2_b, t2, io, B, Cout, H, W, Cout, 3, 3, 1, 1, st);
}

extern "C" void kernel_launch(void* const* d_in, const int* in_sizes, int n_in,
                              void* d_out, int out_size, void* d_ws,
                              size_t ws_size, hipStream_t stream) {
  (void)n_in; (void)out_size; (void)ws_size;
  const float* const* F = (const float* const*)d_in;
  Params P = map_params(F, in_sizes);

  const int B = 8;
  const long ARENA = 67108864;  // 8*128*256*256 elems (268MB)
  float* wsf = (float*)d_ws;
  float* pA = wsf;
  float* pB = wsf + ARENA;
  float* pC = wsf + 2 * ARENA;
  float* stats = wsf + 3 * ARENA;    // 512 floats
  float* sqerr = stats + 512;        // 8192 floats
  float* out = (float*)d_out;        // 24576 quant + 2 losses

  // conv_in: (8,3,256,256) -> (8,64,256,256) in pA
  run_conv(P.x, P.conv_in_w, P.conv_in_b, nullptr, pA, B, 3, 256, 256, 64, 3,
           3, 1, 1, stream);

  // down blocks: resnet (in place) then 4x4 stride-2 downsample
  resnet(P.down[0], pA, pB, pC, B, 64, 128, 256, 256, stats, stream);
  run_conv(pA, P.down[0].ds_w, P.down[0].ds_b, nullptr, pB, B, 128, 256, 256,
           128, 4, 4, 2, 1, stream);  // pB: (8,128,128,128)

  resnet(P.down[1], pB, pA, pC, B, 128, 256, 128, 128, stats, stream);
  run_conv(pB, P.down[1].ds_w, P.down[1].ds_b, nullptr, pA, B, 256, 128, 128,
           256, 4, 4, 2, 1, stream);  // pA: (8,256,64,64)

  resnet(P.down[2], pA, pB, pC, B, 256, 256, 64, 64, stats, stream);
  run_conv(pA, P.down[2].ds_w, P.down[2].ds_b, nullptr, pB, B, 256, 64, 64,
           256, 4, 4, 2, 1, stream);  // pB: (8,256,32,32)

  // mid resnet 1 (in place in pB)
  resnet(P.rn1, pB, pA, pC, B, 256, 256, 32, 32, stats, stream);

  // ---- attention block: input pB, output pC ----
  run_gn(pB, pA, P.agn_s, P.agn_b, B, 256, 1024, 0, stats, stream);  // agn->pA
  // qkv: per batch, M=1024 tokens, N=768, K=256 -> pC (channel-major)
  run_gemm(pA, P.qkv_w, P.qkv_b, nullptr, pC, 1024, 768, 256, 1, 1024, 1, 256,
           1, 1024, B, 1, 256 * 1024, 0, 0, 0, 768 * 1024, 0, 0, 0, 1.f,
           stream);
  // S = (Q K^T)/8 per (b,h): q = pC[ch 0..255], k = pC[ch 256..511]; S -> pA
  float* S = pA;  // agn dead; 33.5M elems
  run_gemm(pC, pC + 256 * 1024, nullptr, nullptr, S, 1024, 1024, 64, 1, 1024,
           1024, 1, 1024, 1, 4, 8, 64 * 1024, 768L * 1024, 64 * 1024,
           768L * 1024, 1024L * 1024, 4L * 1024 * 1024, 0, 0, 0.125f, stream);
  softmax_rows_kernel<<<B * 4 * 1024, 256, 0, stream>>>(S, 1024);
  // O = P V per (b,h) -> obuf (channel-major (8,256,1024)) in upper half of pA
  float* obuf = pA + 33554432;
  run_gemm(S, pC + 512 * 1024, nullptr, nullptr, obuf, 1024, 64, 1024, 1024, 1,
           1, 1024, 1, 1024, 4, 8, 1024L * 1024, 4L * 1024 * 1024, 64 * 1024,
           768L * 1024, 64 * 1024, 256L * 1024, 0, 0, 1.f, stream);
  // out projection + residual(x=pB) -> pC (qkv dead)
  run_gemm(obuf, P.out_w, P.out_b, pB, pC, 1024, 256, 256, 1, 1024, 1, 256, 1,
           1024, B, 1, 256 * 1024, 0, 0, 0, 256 * 1024, 0, 256 * 1024, 0, 1.f,
           stream);

  // mid resnet 2 (in place in pC)
  resnet(P.rn2, pC, pA, pB, B, 256, 256, 32, 32, stats, stream);

  // final norm + silu + conv_out + pq
  run_gn(pC, pA, P.no_s, P.no_b, B, 256, 1024, 1, stats, stream);
  run_conv(pA, P.conv_out_w, P.conv_out_b, nullptr, pB, B, 256, 32, 32, 3, 3,
           3, 1, 1, stream);  // pB: (8,3,32,32)
  run_gemm(pB, P.pq_w, P.pq_b, nullptr, pC, 1024, 3, 3, 1, 1024, 1, 3, 1,
           1024, B, 1, 3 * 1024, 0, 0, 0, 3 * 1024, 0, 0, 0, 1.f,
           stream);  // pC: z_q (8,3,32,32)

  // quantize against codebook; quant -> d_out[0..24575], losses -> [24576..7]
  vq_assign_kernel<<<8192, 256, 0, stream>>>(pC, P.codebook, out, sqerr, 8192);
  vq_loss_kernel<<<1, 256, 0, stream>>>(sqerr, out + 24576, 8192,
                                        1.0f / 24576.0f);
}